// FisherKolmogorovLoss_12841952215404
// MI455X (gfx1250) — compile-verified
//
#include <hip/hip_runtime.h>

// ---------------------------------------------------------------------------
// Fisher-Kolmogorov residual loss: mean( (du_dt - (D*lap(u) + rho*u*(1-u)))^2 )
// Shapes: (4,1,128,128,128) fp32. Bandwidth-bound: 134 MB / 23.3 TB/s ~ 6 us.
// CDNA5 strategy: async global->LDS slice copies (ASYNCcnt path) with a
// depth-3 software pipeline (uniform 2 asyncs/wave/fetch => s_wait_asynccnt 4,
// in-order completion), 8-slice LDS ring, one barrier per z-step, z-register
// chain, deterministic 2-stage reduction. WMMA deliberately unused: no matmul
// structure, the kernel is byte-bound.
// ---------------------------------------------------------------------------

#define NX      128
#define TXDIM   32
#define TYDIM   8
#define HX      (TXDIM + 2)      // 34
#define HY      (TYDIM + 2)      // 10
#define HN      (HX * HY)        // 340 real halo-tile entries per slice
#define SLICE   512              // padded slice stride (2 entries/thread)
#define NBUF    8                // slice ring depth (power of two)
#define NTHR    (TXDIM * TYDIM)  // 256 threads = 8 wave32
#define ZCHUNK  32

// ---- CDNA5 async global->LDS copy ----------------------------------------
#if __has_builtin(__builtin_amdgcn_global_load_async_to_lds_b32)
#define USE_ASYNC_BUILTIN 1
#else
#define USE_ASYNC_BUILTIN 0
#endif

typedef __attribute__((address_space(1))) int* gptr_t;
typedef __attribute__((address_space(3))) int* lptr_t;

__device__ __forceinline__ void async_g2l_b32(const float* g, float* l) {
#if USE_ASYNC_BUILTIN
    __builtin_amdgcn_global_load_async_to_lds_b32((gptr_t)g, (lptr_t)l, 0, 0);
#else
    asm volatile("global_load_async_to_lds_b32 %0, %1, off"
                 :: "v"((unsigned)(unsigned long long)l), "v"(g)
                 : "memory");
#endif
}

__device__ __forceinline__ void wait_async0() {
#if __has_builtin(__builtin_amdgcn_s_wait_asynccnt)
    __builtin_amdgcn_s_wait_asynccnt(0);
#else
    asm volatile("s_wait_asynccnt 0" ::: "memory");
#endif
}
__device__ __forceinline__ void wait_async4() {
#if __has_builtin(__builtin_amdgcn_s_wait_asynccnt)
    __builtin_amdgcn_s_wait_asynccnt(4);
#else
    asm volatile("s_wait_asynccnt 4" ::: "memory");
#endif
}

// ---------------------------------------------------------------------------
__global__ __launch_bounds__(NTHR)
void fk_residual_kernel(const float* __restrict__ u,
                        const float* __restrict__ du_dt,
                        const float* __restrict__ Dv,
                        const float* __restrict__ rho,
                        float* __restrict__ partial)
{
    __shared__ float tile[NBUF][SLICE];   // 16 KB slice ring (with x/y halo)
    __shared__ float wsum[TYDIM];

    const int tx = threadIdx.x, ty = threadIdx.y;
    const int t  = ty * TXDIM + tx;          // 0..255
    const int bx = blockIdx.x;               // 0..3   (x tiles)
    const int by = blockIdx.y;               // 0..15  (y tiles)
    const int bzc = blockIdx.z & 3;          // z chunk
    const int b   = blockIdx.z >> 2;         // batch

    const int x0 = bx * TXDIM, y0 = by * TYDIM, z0 = bzc * ZCHUNK;

    // Halo-entry mapping: entries e0 = t (0..255) and e1 = t+256 (256..511);
    // real entries are e < 340, the rest is write-only padding.
    const int  e0  = t;
    const int  gx0 = x0 + (e0 % HX) - 1, gy0 = y0 + (e0 / HX) - 1;
    const bool in0 = ((unsigned)gx0 < NX) && ((unsigned)gy0 < NX);
    const int  off0 = in0 ? (gy0 * NX + gx0) : 0;      // clamped-safe offset

    const int  e1  = t + NTHR;
    const int  gx1 = x0 + (e1 % HX) - 1, gy1 = y0 + (e1 / HX) - 1;
    const bool real1 = (e1 < HN);
    const bool in1 = real1 && ((unsigned)gx1 < NX) && ((unsigned)gy1 < NX);
    const int  off1 = in1 ? (gy1 * NX + gx1) : 0;

    const size_t plane = (size_t)NX * NX;
    const float* uB = u + (size_t)b * NX * plane;

    // fetch(s): EXACTLY 2 async b32 per lane, every lane, every call ->
    // uniform per-wave ASYNCcnt, enabling depth-3 waits. OOB lanes fetch a
    // clamped-safe address; fix(s) overwrites those entries with 0.
    auto fetch = [&](int s) {
        const int gz = s < 0 ? 0 : (s > NX - 1 ? NX - 1 : s);
        const float* src = uB + (size_t)gz * plane;
        float* dst = tile[(s + 1) & (NBUF - 1)];
        async_g2l_b32(src + off0, dst + e0);
        async_g2l_b32(src + off1, dst + e1);
    };
    auto fix = [&](int s) {
        const bool zok = ((unsigned)s < (unsigned)NX);
        float* dst = tile[(s + 1) & (NBUF - 1)];
        if (!(in0 && zok)) dst[e0] = 0.0f;
        if (real1 && !(in1 && zok)) dst[e1] = 0.0f;
    };

    // Prologue: land slices z0-1, z0; put z0+1 and z0+2 in flight (depth 3).
    fetch(z0 - 1);
    fetch(z0);
    wait_async0();
    fix(z0 - 1);
    fix(z0);
    fetch(z0 + 1);
    fetch(z0 + 2);
    __syncthreads();

    const int gx = x0 + tx, gy = y0 + ty;
    const int c  = (ty + 1) * HX + (tx + 1);

    // z-register chain for the centers of slices z-1 and z.
    float u_zm = tile[(z0)     & (NBUF - 1)][c];
    float u_c  = tile[(z0 + 1) & (NBUF - 1)][c];

    const float* pDu = du_dt + (size_t)b * NX * plane + (size_t)z0 * plane
                     + (size_t)gy * NX + gx;
    const float* pD  = Dv  + (pDu - du_dt);
    const float* pR  = rho + (pDu - du_dt);

    float acc = 0.0f;

    for (int z = z0; z < z0 + ZCHUNK; ++z) {
        fetch(z + 3);        // keep 3 slice-fetch pairs in flight
        const float vdu = *pDu;          // overlap pointwise streams
        const float vD  = *pD;
        const float vR  = *pR;
        pDu += plane; pD += plane; pR += plane;

        wait_async4();       // in-order completion => slice z+1 has landed
        fix(z + 1);          // boundary lanes zero their entries (EXECZ else)
        __syncthreads();     // single barrier: ring depth 8 covers WAR skew
                             // (live slices z-2..z+4 = 7 distinct buffers)

        const float* curr = tile[(z + 1) & (NBUF - 1)];
        const float* next = tile[(z + 2) & (NBUF - 1)];

        const float u_zp = next[c];
        const float lap  = curr[c - 1] + curr[c + 1]
                         + curr[c - HX] + curr[c + HX]
                         + u_zm + u_zp - 6.0f * u_c;
        const float res  = vdu - (vD * lap + vR * u_c * (1.0f - u_c));
        acc += res * res;

        u_zm = u_c;          // rotate z-register chain
        u_c  = u_zp;
    }

    // Deterministic in-block reduction: wave32 shfl tree, then 8 partials.
    for (int off = 16; off > 0; off >>= 1)
        acc += __shfl_xor(acc, off, 32);
    if (tx == 0) wsum[ty] = acc;
    __syncthreads();
    if (t == 0) {
        float s = 0.0f;
#pragma unroll
        for (int i = 0; i < TYDIM; ++i) s += wsum[i];
        const int blin = ((int)blockIdx.z * gridDim.y + blockIdx.y) * gridDim.x
                       + blockIdx.x;
        partial[blin] = s;
    }
}

// Final deterministic reduction of 1024 block partials -> mean.
__global__ __launch_bounds__(1024)
void fk_reduce_kernel(const float* __restrict__ partial,
                      float* __restrict__ out, int n, float scale)
{
    __shared__ float s[32];
    const int t = threadIdx.x;
    float v = (t < n) ? partial[t] : 0.0f;
    for (int off = 16; off > 0; off >>= 1) v += __shfl_xor(v, off, 32);
    if ((t & 31) == 0) s[t >> 5] = v;
    __syncthreads();
    if (t < 32) {
        const int nw = (blockDim.x + 31) >> 5;
        float w = (t < nw) ? s[t] : 0.0f;
        for (int off = 16; off > 0; off >>= 1) w += __shfl_xor(w, off, 32);
        if (t == 0) out[0] = w * scale;
    }
}

// ---------------------------------------------------------------------------
extern "C" void kernel_launch(void* const* d_in, const int* in_sizes, int n_in,
                              void* d_out, int out_size, void* d_ws, size_t ws_size,
                              hipStream_t stream)
{
    const float* u     = (const float*)d_in[0];
    const float* du_dt = (const float*)d_in[1];
    const float* Dv    = (const float*)d_in[2];
    const float* rho   = (const float*)d_in[3];
    float* partial = (float*)d_ws;     // 1024 floats of scratch
    float* out     = (float*)d_out;

    dim3 block(TXDIM, TYDIM, 1);       // 256 threads = 8 wave32
    dim3 grid(NX / TXDIM,              // 4  x-tiles
              NX / TYDIM,              // 16 y-tiles
              (NX / ZCHUNK) * 4);      // 4 z-chunks * 4 batches = 16

    fk_residual_kernel<<<grid, block, 0, stream>>>(u, du_dt, Dv, rho, partial);

    const int nPartials = (NX / TXDIM) * (NX / TYDIM) * (NX / ZCHUNK) * 4; // 1024
    const float scale = 1.0f / (4.0f * 128.0f * 128.0f * 128.0f);
    fk_reduce_kernel<<<1, 1024, 0, stream>>>(partial, out, nPartials, scale);
}